// MPGNN_88459146428944
// MI455X (gfx1250) — compile-verified
//
#include <hip/hip_runtime.h>

typedef __attribute__((ext_vector_type(2))) float v2f;
typedef __attribute__((ext_vector_type(8))) float v8f;

// ---------------------------------------------------------------------------
// WMMA helper: D = A(16x4 f32) * B(4x16 f32) + C(16x16 f32), wave32.
// ---------------------------------------------------------------------------
__device__ __forceinline__ v8f wmma_f32x4(v2f a, v2f b, v8f c) {
    return __builtin_amdgcn_wmma_f32_16x16x4_f32(
        /*neg_a=*/false, a, /*neg_b=*/false, b,
        /*c_mod=*/(short)0, c, /*reuse_a=*/false, /*reuse_b=*/false);
}

__device__ __forceinline__ void atomic_add_f32(float* p, float v) {
#if defined(__HIP_DEVICE_COMPILE__)
    unsafeAtomicAdd(p, v);   // hardware global_atomic_add_f32
#else
    atomicAdd(p, v);
#endif
}

// ---------------------------------------------------------------------------
// LDS-staged GEMM: Y[M,N] = X[M,K] @ W[N,K]^T + bias[N]
// One block (8 waves) per 16-row strip. The strip (16 full rows, ld==K) is one
// contiguous 16*K block -> coalesced float4 copy to LDS, 8x A-reuse across the
// column tiles. Wave w computes tiles (strip, tn = w, w+8, ...).
//   A frag: lane m=(l&15), K pair at koff=2*(l>>4)  (ds_load_b64 from LDS)
//   B frag: lane n=(l&15), B[k,n]=W[n,k]            (contiguous float2)
//   C/D:    v8f, col n=(l&15), rows r + 8*(l>>4)
// ---------------------------------------------------------------------------
template <int K>
__global__ void gemm_bias_wmma_lds(const float* __restrict__ X,
                                   const float* __restrict__ W,
                                   const float* __restrict__ bias,
                                   float* __restrict__ Y,
                                   int M, int N) {
    __shared__ float As[16 * K];

    const int lane = threadIdx.x & 31;
    const int wave = threadIdx.x >> 5;
    const int wavesPerBlock = blockDim.x >> 5;
    const int tilesN = N >> 4;
    const int strip = blockIdx.x * 16;

    // ---- stage A strip to LDS (contiguous when fully in range) ----
    if (strip + 16 <= M) {
        const float4* __restrict__ src = (const float4*)(X + (size_t)strip * K);
        float4* dst = (float4*)As;
        for (int i = threadIdx.x; i < (16 * K) / 4; i += blockDim.x) dst[i] = src[i];
    } else {
        for (int i = threadIdx.x; i < 16 * K; i += blockDim.x) {
            const int rr = i / K, cc = i - rr * K;
            const int gr = min(strip + rr, M - 1);
            As[i] = X[(size_t)gr * K + cc];
        }
    }
    __syncthreads();

    const int koff = (lane >> 4) << 1;
    const float* __restrict__ arow = As + (lane & 15) * K + koff;
    const bool fullTile = (strip + 16 <= M);
    const int rbase = strip + ((lane >> 4) << 3);

    for (int tn = wave; tn < tilesN; tn += wavesPerBlock) {
        const int ncol = tn * 16 + (lane & 15);
        const float* __restrict__ wrow = W + (size_t)ncol * K + koff;

        v8f acc = {};
#pragma unroll
        for (int k = 0; k < K; k += 4) {
            v2f a = *(const v2f*)(arow + k);        // ds_load_b64
            v2f b = *(const v2f*)(wrow + k);        // global_load_b64 (L0-hot)
            acc = wmma_f32x4(a, b, acc);
        }

        const float bv = bias[ncol];
        float* yp = Y + (size_t)rbase * N + ncol;
        if (fullTile) {                              // wave-uniform: branchless stores
#pragma unroll
            for (int r = 0; r < 8; ++r) { *yp = acc[r] + bv; yp += N; }
        } else {
#pragma unroll
            for (int r = 0; r < 8; ++r) {
                if (rbase + r < M) *yp = acc[r] + bv;
                yp += N;
            }
        }
    }
}

// Generic (runtime-K) fallback, no LDS staging.
__global__ void gemm_bias_wmma_rt(const float* __restrict__ X,
                                  const float* __restrict__ W,
                                  const float* __restrict__ bias,
                                  float* __restrict__ Y,
                                  int M, int K, int N) {
    const int lane = threadIdx.x & 31;
    const int wave = threadIdx.x >> 5;
    const int tilesN = N >> 4;
    const int tile = blockIdx.x * (blockDim.x >> 5) + wave;
    const int tm = tile / tilesN;
    const int tn = tile - tm * tilesN;
    if (tm * 16 >= M) return;

    const int mrow = min(tm * 16 + (lane & 15), M - 1);
    const int ncol = tn * 16 + (lane & 15);
    const int koff = (lane >> 4) << 1;
    const float* __restrict__ xrow = X + (size_t)mrow * K + koff;
    const float* __restrict__ wrow = W + (size_t)ncol * K + koff;

    v8f acc = {};
#pragma unroll 8
    for (int k = 0; k < K; k += 4) {
        v2f a = *(const v2f*)(xrow + k);
        v2f b = *(const v2f*)(wrow + k);
        acc = wmma_f32x4(a, b, acc);
    }

    const float bv = bias[ncol];
    const int rbase = tm * 16 + ((lane >> 4) << 3);
    float* yp = Y + (size_t)rbase * N + ncol;
#pragma unroll
    for (int r = 0; r < 8; ++r) {
        if (rbase + r < M) *yp = acc[r] + bv;
        yp += N;
    }
}

// ---------------------------------------------------------------------------
// Edge MLP: out[e, 0:C] = concat(x[row_e], x[Nu+col_e]) @ We[C,2H]^T + be
// One wave per 16 edges; output cols padded 10 -> 16 via multiply-mask.
// ---------------------------------------------------------------------------
template <int HT>
__global__ void edge_mlp_wmma(const float* __restrict__ xf,
                              const int* __restrict__ row,
                              const int* __restrict__ col,
                              const float* __restrict__ We,
                              const float* __restrict__ be,
                              float* __restrict__ out,
                              int E, int Nu, int Hrt, int C) {
    const int H = (HT > 0) ? HT : Hrt;
    const int K = 2 * H;
    const int lane = threadIdx.x & 31;
    const int wave = threadIdx.x >> 5;
    const int tile = blockIdx.x * (blockDim.x >> 5) + wave;
    const int ebase = tile * 16;
    if (ebase >= E) return;                   // wave-uniform

    const int m = lane & 15;
    const int e = min(ebase + m, E - 1);
    const int r = row[e];
    const int c = col[e];
    const int koff = (lane >> 4) << 1;

    const float* __restrict__ urow = xf + (size_t)r * H + koff;
    const float* __restrict__ mrow = xf + (size_t)(Nu + c) * H + koff;

    const int n = lane & 15;
    const float wmask = (n < C) ? 1.0f : 0.0f;          // zero-pad cols C..15
    const float* __restrict__ wrow = We + (size_t)min(n, C - 1) * K + koff;

    v8f acc = {};
#pragma unroll 8
    for (int k = 0; k < K; k += 4) {
        const float* src = (k + koff < H) ? (urow + k) : (mrow + (k - H));
        v2f a = *(const v2f*)src;
        v2f b = *(const v2f*)(wrow + k);
        b.x *= wmask;
        b.y *= wmask;
        acc = wmma_f32x4(a, b, acc);
    }

    if (n < C) {
        const float bv = be[n];
        const int mb = (lane >> 4) << 3;
        float* op = out + (size_t)(ebase + mb) * C + n;
        if (ebase + 16 <= E) {                           // wave-uniform fast path
#pragma unroll
            for (int rr = 0; rr < 8; ++rr) { *op = acc[rr] + bv; op += C; }
        } else {
#pragma unroll
            for (int rr = 0; rr < 8; ++rr) {
                if (ebase + mb + rr < E) *op = acc[rr] + bv;
                op += C;
            }
        }
    }
}

// ---------------------------------------------------------------------------
// Graph-aggregation support kernels
// ---------------------------------------------------------------------------
__global__ void fill_kernel(float* __restrict__ p, float v, long long n) {
    long long i = (long long)blockIdx.x * blockDim.x + threadIdx.x;
    if (i < n) p[i] = v;
}

__global__ void zero4_kernel(float4* __restrict__ p, long long n4) {
    long long i = (long long)blockIdx.x * blockDim.x + threadIdx.x;
    if (i < n4) p[i] = make_float4(0.f, 0.f, 0.f, 0.f);
}

__global__ void deg_scatter_kernel(const int* __restrict__ col,
                                   float* __restrict__ deg, int E) {
    int e = blockIdx.x * blockDim.x + threadIdx.x;
    if (e < E) atomic_add_f32(&deg[col[e]], 1.0f);
}

__global__ void rsqrt_kernel(float* __restrict__ d, int N) {
    int i = blockIdx.x * blockDim.x + threadIdx.x;
    if (i < N) d[i] = rsqrtf(d[i]);
}

// per edge: acc[col] += dinv[row] * h[row]   (32 lanes x float4 per edge, H=128)
__global__ void edge_scatter_kernel(const float* __restrict__ h,
                                    const float* __restrict__ dinv,
                                    const int* __restrict__ row,
                                    const int* __restrict__ col,
                                    float* __restrict__ acc,
                                    int E, int H) {
    const int chunks = H >> 2;                    // float4 chunks per row
    long long gid = (long long)blockIdx.x * blockDim.x + threadIdx.x;
    long long total = (long long)E * chunks;
    if (gid >= total) return;
    const int e = (int)(gid / chunks);
    const int j = (int)(gid - (long long)e * chunks) << 2;
    const int r = row[e];
    const int c = col[e];
    const float s = dinv[r];
    const float4 hv = *(const float4*)(h + (size_t)r * H + j);
    float* dst = acc + (size_t)c * H + j;
    atomic_add_f32(dst + 0, s * hv.x);
    atomic_add_f32(dst + 1, s * hv.y);
    atomic_add_f32(dst + 2, s * hv.z);
    atomic_add_f32(dst + 3, s * hv.w);
}

// x[i,j] = relu(dinv[i]*acc[i,j] + dinv[i]^2*h[i,j]); grid = N nodes, block = H
__global__ void finalize_relu_kernel(const float* __restrict__ h,
                                     const float* __restrict__ dinv,
                                     float* __restrict__ acc_inout, int H) {
    const int node = blockIdx.x;
    const int j = threadIdx.x;
    const float di = dinv[node];
    const size_t idx = (size_t)node * H + j;
    const float v = di * acc_inout[idx] + di * di * h[idx];
    acc_inout[idx] = v > 0.f ? v : 0.f;
}

// ---------------------------------------------------------------------------
// Host launch
// ---------------------------------------------------------------------------
static inline int ceil_div_i(long long a, long long b) { return (int)((a + b - 1) / b); }

static void launch_gemm(const float* X, const float* W, const float* b, float* Y,
                        int M, int K, int N, hipStream_t stream) {
    const int TPB = 256;
    const int strips = ceil_div_i(M, 16);
    if (K == 32)
        gemm_bias_wmma_lds<32><<<strips, TPB, 0, stream>>>(X, W, b, Y, M, N);
    else if (K == 64)
        gemm_bias_wmma_lds<64><<<strips, TPB, 0, stream>>>(X, W, b, Y, M, N);
    else if (K == 128)
        gemm_bias_wmma_lds<128><<<strips, TPB, 0, stream>>>(X, W, b, Y, M, N);
    else {
        const int tiles = strips * (N / 16);
        gemm_bias_wmma_rt<<<ceil_div_i(tiles, 8), TPB, 0, stream>>>(X, W, b, Y, M, K, N);
    }
}

extern "C" void kernel_launch(void* const* d_in, const int* in_sizes, int n_in,
                              void* d_out, int out_size, void* d_ws, size_t ws_size,
                              hipStream_t stream) {
    const float* x_user  = (const float*)d_in[0];
    const float* x_movie = (const float*)d_in[1];
    const int*   edge    = (const int*)d_in[2];
    const float* Wu = (const float*)d_in[3];
    const float* bu = (const float*)d_in[4];
    const float* Wm = (const float*)d_in[5];
    const float* bm = (const float*)d_in[6];
    const float* W1 = (const float*)d_in[7];
    const float* b1 = (const float*)d_in[8];
    const float* W2 = (const float*)d_in[9];
    const float* b2 = (const float*)d_in[10];
    const float* We = (const float*)d_in[11];
    const float* be = (const float*)d_in[12];
    float* out = (float*)d_out;

    const int H  = in_sizes[4];            // 128
    const int C  = in_sizes[12];           // 10
    const int Fu = in_sizes[3] / H;        // 32
    const int Fm = in_sizes[5] / H;        // 64
    const int Nu = in_sizes[0] / Fu;       // 50000
    const int Nm = in_sizes[1] / Fm;       // 50000
    const int E  = in_sizes[2] / 2;        // 1e6
    const int N  = Nu + Nm;                // 100000
    const int* row = edge;
    const int* col = edge + E;

    float* ws   = (float*)d_ws;
    float* X    = ws;                                  // [N, H]
    float* Hh   = X  + (size_t)N * H;                  // [N, H]
    float* Acc  = Hh + (size_t)N * H;                  // [N, H]
    float* dinv = Acc + (size_t)N * H;                 // [N]

    const int TPB = 256, WPB = 8;
    const long long NH = (long long)N * H;

    // 1. input transforms -> X
    launch_gemm(x_user, Wu, bu, X, Nu, Fu, H, stream);
    launch_gemm(x_movie, Wm, bm, X + (size_t)Nu * H, Nm, Fm, H, stream);

    // 2. degrees (self-loop = init 1.0) -> dinv = deg^-1/2
    fill_kernel<<<ceil_div_i(N, TPB), TPB, 0, stream>>>(dinv, 1.0f, N);
    deg_scatter_kernel<<<ceil_div_i(E, TPB), TPB, 0, stream>>>(col, dinv, E);
    rsqrt_kernel<<<ceil_div_i(N, TPB), TPB, 0, stream>>>(dinv, N);

    const long long scatterThreads = (long long)E * (H / 4);

    // 3. conv1: h = X@W1^T+b1 ; Acc = relu-normalized aggregate
    launch_gemm(X, W1, b1, Hh, N, H, H, stream);
    zero4_kernel<<<ceil_div_i(NH / 4, TPB), TPB, 0, stream>>>((float4*)Acc, NH / 4);
    edge_scatter_kernel<<<ceil_div_i(scatterThreads, TPB), TPB, 0, stream>>>(Hh, dinv, row, col, Acc, E, H);
    finalize_relu_kernel<<<N, H, 0, stream>>>(Hh, dinv, Acc, H);

    // 4. conv2: h = Acc@W2^T+b2 ; X = relu-normalized aggregate
    launch_gemm(Acc, W2, b2, Hh, N, H, H, stream);
    zero4_kernel<<<ceil_div_i(NH / 4, TPB), TPB, 0, stream>>>((float4*)X, NH / 4);
    edge_scatter_kernel<<<ceil_div_i(scatterThreads, TPB), TPB, 0, stream>>>(Hh, dinv, row, col, X, E, H);
    finalize_relu_kernel<<<N, H, 0, stream>>>(Hh, dinv, X, H);

    // 5. edge MLP -> out [E, C]
    const int edgeTiles = ceil_div_i(E, 16);
    if (H == 128)
        edge_mlp_wmma<128><<<ceil_div_i(edgeTiles, WPB), TPB, 0, stream>>>(X, row, col, We, be, out, E, Nu, H, C);
    else
        edge_mlp_wmma<0><<<ceil_div_i(edgeTiles, WPB), TPB, 0, stream>>>(X, row, col, We, be, out, E, Nu, H, C);
}